// Encoder_89352499626364
// MI455X (gfx1250) — compile-verified
//
#include <hip/hip_runtime.h>
#include <hip/hip_bf16.h>

typedef float v2f __attribute__((ext_vector_type(2)));
typedef float v8f __attribute__((ext_vector_type(8)));

#define D_H 128   // output width of both layers
#define KC  32    // K-chunk staged in LDS per iteration
#define AS_STRIDE 34   // 32 + 2 pad floats (bank-conflict avoidance)
#define BS_STRIDE 130  // 128 + 2 pad floats

// ---- CDNA5 async global->LDS copies (CDNA5 ISA) ----------
__device__ __forceinline__ void async_g2l_b64(unsigned ldsOff, const float* g) {
  asm volatile("global_load_async_to_lds_b64 %0, %1, off"
               :: "v"(ldsOff), "v"(g) : "memory");
}
__device__ __forceinline__ void async_g2l_b128(unsigned ldsOff, const float* g) {
  asm volatile("global_load_async_to_lds_b128 %0, %1, off"
               :: "v"(ldsOff), "v"(g) : "memory");
}
__device__ __forceinline__ void wait_async0() {
  asm volatile("s_wait_asynccnt 0x0" ::: "memory");
}
__device__ __forceinline__ unsigned lds_off(const void* p) {
  return (unsigned)(unsigned long long)p;  // flat LDS aperture: low 32 bits
}

// ---------------- degree / normalization ----------------

__global__ void k_deg_init(float* __restrict__ deg, int n) {
  int i = blockIdx.x * blockDim.x + threadIdx.x;
  if (i < n) deg[i] = 1.0f;  // self-loop contributes 1
}

__global__ void k_deg_count(const long long* __restrict__ dst,
                            float* __restrict__ deg, int E) {
  int e = blockIdx.x * blockDim.x + threadIdx.x;
  if (e < E) unsafeAtomicAdd(&deg[(int)dst[e]], 1.0f);
}

__global__ void k_rsqrt(const float* __restrict__ deg,
                        float* __restrict__ dis, int n) {
  int i = blockIdx.x * blockDim.x + threadIdx.x;
  if (i < n) dis[i] = rsqrtf(deg[i]);  // deg >= 1 always
}

__global__ void k_zero(float* __restrict__ p, size_t n) {
  size_t i = (size_t)blockIdx.x * blockDim.x + threadIdx.x;
  if (i < n) p[i] = 0.0f;
}

// ---------------- dense GEMM: H[n,128] = X[n,K] @ W[K,128] -----------------
// 256 threads = 8 waves; block computes a 16-row x 128-col strip; wave w owns
// the 16x16 tile at cols [w*16,+16). K consumed in KC=32 chunks, double-
// buffered in LDS: chunk c+1's async global->LDS copies overlap chunk c's
// 8 x V_WMMA_F32_16X16X4_F32.
__global__ void __launch_bounds__(256)
k_gemm_wmma(const float* __restrict__ X, const float* __restrict__ W,
            float* __restrict__ H, int n, int K) {
  __shared__ float As[2][16 * AS_STRIDE];
  __shared__ float Bs[2][KC * BS_STRIDE];

  const int t    = threadIdx.x;
  const int lane = t & 31;
  const int wave = t >> 5;
  const int half = lane >> 4;        // 0: K=k,k+1   1: K=k+2,k+3
  const int lm   = lane & 15;
  const int row0 = blockIdx.x * 16;
  const int col  = wave * 16 + lm;

  // Cooperative-load coordinates (branch-free: clamp row instead of masking)
  const int am = t >> 4;                       // A row 0..15
  const int ak = (t & 15) * 2;                 // A k-offset 0,2,..,30
  const int arow = min(row0 + am, n - 1);
  const float* aG = X + (size_t)arow * K + ak;
  const int br = t >> 5;                       // B row base for this thread
  const int bc = (t & 31) * 4;                 // B col 0,4,..,124

  auto issue = [&](int k0, int buf) {
    async_g2l_b64(lds_off(&As[buf][am * AS_STRIDE + ak]), aG + k0);
    #pragma unroll
    for (int it = 0; it < 4; ++it) {           // 32 rows = 4 x 8 row-groups
      const int r = it * 8 + br;
      async_g2l_b128(lds_off(&Bs[buf][r * BS_STRIDE + bc]),
                     W + (size_t)(k0 + r) * D_H + bc);
    }
  };

  v8f acc = {};
  const int NC = K / KC;
  issue(0, 0);
  wait_async0();
  __syncthreads();

  for (int c = 0; c < NC; ++c) {
    if (c + 1 < NC) issue((c + 1) * KC, (c + 1) & 1);   // overlap with compute

    const float* Asb = As[c & 1];
    const float* Bsb = Bs[c & 1];
    #pragma unroll
    for (int kk = 0; kk < KC; kk += 4) {
      const int ka = kk + 2 * half;
      v2f a = *(const v2f*)&Asb[lm * AS_STRIDE + ka];    // ds_load_b64
      v2f b;
      b.x = Bsb[ka * BS_STRIDE + col];
      b.y = Bsb[(ka + 1) * BS_STRIDE + col];
      acc = __builtin_amdgcn_wmma_f32_16x16x4_f32(
          false, a, false, b, (short)0, acc, false, false);
    }

    wait_async0();      // chunk c+1 landed (copies ran during the WMMAs above)
    __syncthreads();    // all waves done reading buf[c&1]; safe to reuse
  }

  // Epilogue: block-uniform fast path avoids per-lane exec masking
  const int rbase = row0 + half * 8;           // C/D: VGPR rr -> M = 8*half + rr
  if (row0 + 16 <= n) {
    float* o = H + (size_t)rbase * D_H + col;
    #pragma unroll
    for (int rr = 0; rr < 8; ++rr) o[(size_t)rr * D_H] = acc[rr];
  } else {
    #pragma unroll
    for (int rr = 0; rr < 8; ++rr)
      if (rbase + rr < n) H[(size_t)(rbase + rr) * D_H + col] = acc[rr];
  }
}

// ---------------- edge scatter: AGG[dst] += H[src] * dis[s]*dis[d] ---------
__global__ void k_scatter(const long long* __restrict__ src,
                          const long long* __restrict__ dst,
                          const float* __restrict__ dis,
                          const float* __restrict__ H,
                          float* __restrict__ AGG, int E) {
  const int e = blockIdx.x * (blockDim.x >> 5) + (threadIdx.x >> 5);
  if (e >= E) return;
  const int lane = threadIdx.x & 31;
  const int s = (int)src[e];
  const int d = (int)dst[e];
  const float norm = dis[s] * dis[d];
  const float4 v = ((const float4*)(H + (size_t)s * D_H))[lane];
  float* out = AGG + (size_t)d * D_H + lane * 4;
  unsafeAtomicAdd(out + 0, v.x * norm);
  unsafeAtomicAdd(out + 1, v.y * norm);
  unsafeAtomicAdd(out + 2, v.z * norm);
  unsafeAtomicAdd(out + 3, v.w * norm);
}

// ---------------- finalize: OUT += H*dis^2 (self-loop) + bias, opt ReLU ----
__global__ void k_finalize(const float* __restrict__ H,
                           const float* __restrict__ dis,
                           const float* __restrict__ bias,
                           float* __restrict__ OUT, int n, int relu) {
  const size_t idx = (size_t)blockIdx.x * blockDim.x + threadIdx.x;
  if (idx >= (size_t)n * D_H) return;
  const int i = (int)(idx >> 7);
  const int j = (int)(idx & 127);
  const float di = dis[i];
  float v = OUT[idx] + H[idx] * di * di + bias[j];
  OUT[idx] = relu ? fmaxf(v, 0.0f) : v;
}

// ---------------- host side ----------------

extern "C" void kernel_launch(void* const* d_in, const int* in_sizes, int n_in,
                              void* d_out, int out_size, void* d_ws, size_t ws_size,
                              hipStream_t stream) {
  const float*     x   = (const float*)d_in[0];      // [n, 256]
  const long long* ei  = (const long long*)d_in[1];  // [2, E] int64
  const float*     W1  = (const float*)d_in[2];      // [256, 128]
  const float*     b1  = (const float*)d_in[3];      // [128]
  const float*     W2  = (const float*)d_in[4];      // [128, 128]
  const float*     b2  = (const float*)d_in[5];      // [128]
  float*           out = (float*)d_out;              // [n, 128]

  const int n = in_sizes[0] / 256;
  const int E = in_sizes[1] / 2;
  const long long* srcp = ei;
  const long long* dstp = ei + E;

  // Workspace: deg[n] | dis[n] | h1[n*128] | x1[n*128] | h2[n*128]
  float* ws  = (float*)d_ws;
  float* deg = ws;
  float* dis = deg + n;
  float* h1  = dis + n;
  float* x1  = h1 + (size_t)n * D_H;
  float* h2  = x1 + (size_t)n * D_H;

  const int T = 256;
  const size_t nf = (size_t)n * D_H;
  const int gN  = (n + T - 1) / T;
  const int gE  = (E + T - 1) / T;
  const int gNF = (int)((nf + T - 1) / T);
  const int gEw = (E + 7) / 8;          // 8 edge-waves per 256-thread block
  const int mtiles = (n + 15) / 16;

  // Normalization
  k_deg_init <<<gN, T, 0, stream>>>(deg, n);
  k_deg_count<<<gE, T, 0, stream>>>(dstp, deg, E);
  k_rsqrt    <<<gN, T, 0, stream>>>(deg, dis, n);

  // Layer 1
  k_gemm_wmma<<<mtiles, 256, 0, stream>>>(x, W1, h1, n, 256);
  k_zero     <<<gNF, T, 0, stream>>>(x1, nf);
  k_scatter  <<<gEw, 256, 0, stream>>>(srcp, dstp, dis, h1, x1, E);
  k_finalize <<<gNF, T, 0, stream>>>(h1, dis, b1, x1, n, 1);

  // Layer 2
  k_gemm_wmma<<<mtiles, 256, 0, stream>>>(x1, W2, h2, n, 128);
  k_zero     <<<gNF, T, 0, stream>>>(out, nf);
  k_scatter  <<<gEw, 256, 0, stream>>>(srcp, dstp, dis, h2, out, E);
  k_finalize <<<gNF, T, 0, stream>>>(h2, dis, b2, out, n, 0);
}